// GCN_12945031431008
// MI455X (gfx1250) — compile-verified
//
#include <hip/hip_runtime.h>

typedef __attribute__((ext_vector_type(2))) float v2f;
typedef __attribute__((ext_vector_type(8))) float v8f;

#if defined(__has_builtin)
#if __has_builtin(__builtin_amdgcn_global_load_async_to_lds_b128) && \
    __has_builtin(__builtin_amdgcn_s_wait_asynccnt)
#define USE_ASYNC_LDS 1
#endif
#endif

#ifdef USE_ASYNC_LDS
typedef int v4i __attribute__((vector_size(4 * sizeof(int))));
typedef __attribute__((address_space(1))) v4i* v4i_glob;
typedef __attribute__((address_space(3))) v4i* v4i_lds;
#endif

// ---------------------------------------------------------------------------
// degree / normalization helpers
// ---------------------------------------------------------------------------
__global__ void fill_f32(float* __restrict__ p, float v, int n) {
  int i = blockIdx.x * blockDim.x + threadIdx.x;
  if (i < n) p[i] = v;
}

__global__ void deg_count(float* __restrict__ deg, const int* __restrict__ dst, int nE) {
  int e = blockIdx.x * blockDim.x + threadIdx.x;
  if (e < nE) atomicAdd(&deg[dst[e]], 1.0f);
}

__global__ void rsqrt_inplace(float* __restrict__ p, int n) {
  int i = blockIdx.x * blockDim.x + threadIdx.x;
  if (i < n) p[i] = rsqrtf(p[i]);  // deg >= 1 always (self-loop)
}

// ---------------------------------------------------------------------------
// WMMA f32 GEMM (full blocks only, NO bounds checks => EXEC stays all-ones
// and no divergent guards around v_wmma):
//   C[M,NDIM] = A[M,KDIM] @ B[KDIM,NDIM], row-major.
// Block = 256 threads = 8 waves. B staged into LDS once per block
// (async-to-LDS), then each wave computes MW consecutive 16x16 M-tiles for
// one N-tile, reusing each B fragment across MW WMMAs.
//
// V_WMMA_F32_16X16X4_F32 fragment layout:
//   A (16x4): lanes 0-15 rows, VGPRs {K=k0,k0+1}; lanes 16-31 {K=k0+2,k0+3}
//   B (4x16): lanes 0-15 cols, VGPRs {K=k0,k0+1}; lanes 16-31 {K=k0+2,k0+3}
//   C/D: VGPR r -> row (half*8 + r), col = lane%16
// ---------------------------------------------------------------------------
template <int NDIM, int KDIM, int MW>
__global__ __launch_bounds__(256) void wmma_gemm_f32(const float* __restrict__ A,
                                                     const float* __restrict__ Bg,
                                                     float* __restrict__ C) {
  constexpr int NT = NDIM / 16;   // N tiles per row strip
  constexpr int MG = 8 / NT;      // M groups (of MW tiles) per block
  constexpr int ROWS_PER_BLOCK = MG * MW * 16;
  __shared__ float ldsB[KDIM * NDIM];

  // ---- stage B into LDS (whole matrix: KDIM*NDIM floats) ----
#ifdef USE_ASYNC_LDS
  for (int i = threadIdx.x * 4; i < KDIM * NDIM; i += 256 * 4) {
    __builtin_amdgcn_global_load_async_to_lds_b128(
        (v4i_glob)(Bg + i), (v4i_lds)(ldsB + i), /*offset=*/0, /*cpol=*/0);
  }
  __builtin_amdgcn_s_wait_asynccnt(0);
  __syncthreads();
#else
  for (int i = threadIdx.x * 4; i < KDIM * NDIM; i += 256 * 4)
    *(float4*)(ldsB + i) = *(const float4*)(Bg + i);
  __syncthreads();
#endif

  const int wave = threadIdx.x >> 5;
  const int lane = threadIdx.x & 31;
  const int nt   = wave % NT;
  const int mg   = wave / NT;
  const int half = lane >> 4;     // selects K pair {k0,k0+1} vs {k0+2,k0+3}
  const int l    = lane & 15;
  const int m0   = blockIdx.x * ROWS_PER_BLOCK + mg * (MW * 16);

  const float* arow[MW];
#pragma unroll
  for (int t = 0; t < MW; ++t)
    arow[t] = A + (size_t)(m0 + t * 16 + l) * KDIM;

  v8f acc[MW];
#pragma unroll
  for (int t = 0; t < MW; ++t) acc[t] = (v8f){};

  const float* bcol = ldsB + nt * 16 + l;
#pragma unroll
  for (int kk = 0; kk < KDIM; kk += 4) {
    const int k0 = kk + half * 2;  // even -> 8B-aligned b64 A load
    v2f b;
    b[0] = bcol[k0 * NDIM];
    b[1] = bcol[(k0 + 1) * NDIM];
#pragma unroll
    for (int t = 0; t < MW; ++t) {
      v2f a = *(const v2f*)(arow[t] + k0);
      acc[t] = __builtin_amdgcn_wmma_f32_16x16x4_f32(
          /*neg_a=*/false, a, /*neg_b=*/false, b,
          /*c_mod=*/(short)0, acc[t], /*reuse_a=*/false, /*reuse_b=*/false);
    }
  }

#pragma unroll
  for (int t = 0; t < MW; ++t) {
    float* crow = C + (size_t)(m0 + t * 16 + half * 8) * NDIM + nt * 16 + l;
#pragma unroll
    for (int r = 0; r < 8; ++r) crow[(size_t)r * NDIM] = acc[t][r];
  }
}

// ---------------------------------------------------------------------------
// scalar tail GEMM for the last (M - mStart) rows (tiny: <= a few thousand
// output elements). One thread per output element.
// ---------------------------------------------------------------------------
__global__ void gemm_tail(const float* __restrict__ A, const float* __restrict__ B,
                          float* __restrict__ C, int mStart, int M, int Ncols, int K) {
  int t = blockIdx.x * blockDim.x + threadIdx.x;
  int row = mStart + t / Ncols;
  int col = t % Ncols;
  if (row >= M) return;
  float acc = 0.0f;
  for (int k = 0; k < K; ++k) acc += A[(size_t)row * K + k] * B[(size_t)k * Ncols + col];
  C[(size_t)row * Ncols + col] = acc;
}

// ---------------------------------------------------------------------------
// agg[i,:] = h[i,:] * dinv[i]^2   (self-loop contribution, also zero-inits agg)
// ---------------------------------------------------------------------------
__global__ void selfloop_scale(const float* __restrict__ h, const float* __restrict__ dinv,
                               float* __restrict__ out, int n, int C, int lgPer) {
  long long t = (long long)blockIdx.x * blockDim.x + threadIdx.x;
  int per = 1 << lgPer;  // float4 chunks per node = C/4
  long long i = t >> lgPer;
  int q = (int)(t & (per - 1));
  if (i >= n) return;
  float di = dinv[i];
  float s = di * di;
  float4 v = ((const float4*)(h + (size_t)i * C))[q];
  float4 o = make_float4(v.x * s, v.y * s, v.z * s, v.w * s);
  ((float4*)(out + (size_t)i * C))[q] = o;
}

// ---------------------------------------------------------------------------
// edge scatter: out[dst,:] += h[src,:] * dinv[src]*dinv[dst]
// (C/4) threads per edge, coalesced float4 gather + 4 f32 atomics.
// h and out are L2-resident (<= 25.6 MB each vs 192 MB L2).
// ---------------------------------------------------------------------------
__global__ void edge_scatter(const float* __restrict__ h, const int* __restrict__ src,
                             const int* __restrict__ dst, const float* __restrict__ dinv,
                             float* __restrict__ out, int nE, int C, int lgPer) {
  long long t = (long long)blockIdx.x * blockDim.x + threadIdx.x;
  int per = 1 << lgPer;
  long long e = t >> lgPer;
  int q = (int)(t & (per - 1));
  if (e >= nE) return;
  int s = src[e];
  int d = dst[e];
  float norm = dinv[s] * dinv[d];
  float4 v = ((const float4*)(h + (size_t)s * C))[q];
  float* op = out + (size_t)d * C + q * 4;
  atomicAdd(op + 0, v.x * norm);
  atomicAdd(op + 1, v.y * norm);
  atomicAdd(op + 2, v.z * norm);
  atomicAdd(op + 3, v.w * norm);
}

// ---------------------------------------------------------------------------
// a[i,c] = relu(a[i,c] + b[c])   (in place, float4 granularity)
// ---------------------------------------------------------------------------
__global__ void bias_relu(float* __restrict__ a, const float* __restrict__ b,
                          int n, int C, int lgPer) {
  long long t = (long long)blockIdx.x * blockDim.x + threadIdx.x;
  int per = 1 << lgPer;
  long long i = t >> lgPer;
  int q = (int)(t & (per - 1));
  if (i >= n) return;
  float4 v = ((float4*)(a + (size_t)i * C))[q];
  float4 bb = ((const float4*)b)[q];
  v.x = fmaxf(v.x + bb.x, 0.0f);
  v.y = fmaxf(v.y + bb.y, 0.0f);
  v.z = fmaxf(v.z + bb.z, 0.0f);
  v.w = fmaxf(v.w + bb.w, 0.0f);
  ((float4*)(a + (size_t)i * C))[q] = v;
}

// ---------------------------------------------------------------------------
// out[i,:] = log_softmax(agg[i,:] + b[:])  for 16 channels, 1 thread per node
// ---------------------------------------------------------------------------
__global__ void logsoftmax16(const float* __restrict__ agg, const float* __restrict__ b,
                             float* __restrict__ out, int n) {
  int i = blockIdx.x * blockDim.x + threadIdx.x;
  if (i >= n) return;
  float z[16];
  float m = -3.402823466e38f;
#pragma unroll
  for (int q = 0; q < 4; ++q) {
    float4 v = ((const float4*)(agg + (size_t)i * 16))[q];
    float4 bb = ((const float4*)b)[q];
    z[q * 4 + 0] = v.x + bb.x;
    z[q * 4 + 1] = v.y + bb.y;
    z[q * 4 + 2] = v.z + bb.z;
    z[q * 4 + 3] = v.w + bb.w;
  }
#pragma unroll
  for (int c = 0; c < 16; ++c) m = fmaxf(m, z[c]);
  float s = 0.0f;
#pragma unroll
  for (int c = 0; c < 16; ++c) s += __expf(z[c] - m);
  float lse = m + __logf(s);
#pragma unroll
  for (int q = 0; q < 4; ++q) {
    float4 o = make_float4(z[q * 4 + 0] - lse, z[q * 4 + 1] - lse,
                           z[q * 4 + 2] - lse, z[q * 4 + 3] - lse);
    ((float4*)(out + (size_t)i * 16))[q] = o;
  }
}

// ---------------------------------------------------------------------------
extern "C" void kernel_launch(void* const* d_in, const int* in_sizes, int n_in,
                              void* d_out, int out_size, void* d_ws, size_t ws_size,
                              hipStream_t stream) {
  const float* x  = (const float*)d_in[0];
  const float* W1 = (const float*)d_in[1];
  const float* b1 = (const float*)d_in[2];
  const float* W2 = (const float*)d_in[3];
  const float* b2 = (const float*)d_in[4];
  const int*   ei = (const int*)d_in[5];

  const int IN = 64, HID = 64, OUT = 16;
  const int N = in_sizes[0] / IN;   // 100000
  const int E = in_sizes[5] / 2;    // 1000000
  const int* src = ei;
  const int* dst = ei + E;

  float* ws = (float*)d_ws;
  size_t off = 0;
  float* dinv = ws + off; off += (((size_t)N + 63) & ~(size_t)63);
  float* h1   = ws + off; off += (size_t)N * HID;
  float* agg1 = ws + off; off += (size_t)N * HID;
  float* h2   = ws + off; off += (size_t)N * OUT;
  float* agg2 = ws + off;
  float* out  = (float*)d_out;

  const int B = 256;  // 8 waves per block (wave32)

  // degree -> dinv = rsqrt(deg), deg includes self-loop
  fill_f32<<<(N + B - 1) / B, B, 0, stream>>>(dinv, 1.0f, N);
  deg_count<<<(E + B - 1) / B, B, 0, stream>>>(dinv, dst, E);
  rsqrt_inplace<<<(N + B - 1) / B, B, 0, stream>>>(dinv, N);

  // ---- layer 1: h1 = x @ W1 ----
  {
    constexpr int ROWS = 2 * 4 * 16;  // MG=2, MW=4 -> 128 rows / block
    int fullBlocks = N / ROWS;        // 781 -> covers 99968 rows, guard-free
    int mStart = fullBlocks * ROWS;
    wmma_gemm_f32<64, 64, 4><<<fullBlocks, B, 0, stream>>>(x, W1, h1);
    int tailElems = (N - mStart) * HID;  // 32*64 = 2048
    if (tailElems > 0)
      gemm_tail<<<(tailElems + B - 1) / B, B, 0, stream>>>(x, W1, h1, mStart, N, HID, IN);
  }
  {
    long long t = (long long)N * (HID / 4);
    selfloop_scale<<<(int)((t + B - 1) / B), B, 0, stream>>>(h1, dinv, agg1, N, HID, 4);
  }
  {
    long long t = (long long)E * (HID / 4);  // 16 threads / edge
    edge_scatter<<<(int)((t + B - 1) / B), B, 0, stream>>>(h1, src, dst, dinv, agg1, E, HID, 4);
  }
  {
    long long t = (long long)N * (HID / 4);
    bias_relu<<<(int)((t + B - 1) / B), B, 0, stream>>>(agg1, b1, N, HID, 4);
  }

  // ---- layer 2: h2 = relu_agg1 @ W2 ----
  {
    constexpr int ROWS = 8 * 4 * 16;  // MG=8, MW=4 -> 512 rows / block
    int fullBlocks = N / ROWS;        // 195 -> covers 99840 rows, guard-free
    int mStart = fullBlocks * ROWS;
    wmma_gemm_f32<16, 64, 4><<<fullBlocks, B, 0, stream>>>(agg1, W2, h2);
    int tailElems = (N - mStart) * OUT;  // 160*16 = 2560
    if (tailElems > 0)
      gemm_tail<<<(tailElems + B - 1) / B, B, 0, stream>>>(agg1, W2, h2, mStart, N, OUT, HID);
  }
  {
    long long t = (long long)N * (OUT / 4);
    selfloop_scale<<<(int)((t + B - 1) / B), B, 0, stream>>>(h2, dinv, agg2, N, OUT, 2);
  }
  {
    long long t = (long long)E * (OUT / 4);  // 4 threads / edge
    edge_scatter<<<(int)((t + B - 1) / B), B, 0, stream>>>(h2, src, dst, dinv, agg2, E, OUT, 2);
  }
  logsoftmax16<<<(N + B - 1) / B, B, 0, stream>>>(agg2, b2, out, N);
}